// Quantizer_39281770889521
// MI455X (gfx1250) — compile-verified
//
#include <hip/hip_runtime.h>
#include <hip/hip_fp16.h>
#include <float.h>
#include <math.h>

// LAC int4 symmetric row quantizer for MI455X (gfx1250, wave32).
// Bandwidth-bound streaming kernel: 1 read + 1 write of x (~256 MiB total,
// ~11 us floor at 23.3 TB/s). Row is staged global->LDS with the CDNA5
// async data-mover path (GLOBAL_LOAD_ASYNC_TO_LDS_B128 + s_wait_asynccnt).

#define ROW_D  4096
#define BLOCK  256
#define V4PT   (ROW_D / (BLOCK * 4))   // float4 chunks per thread = 4
#define NWAVES (BLOCK / 32)

typedef float f4 __attribute__((ext_vector_type(4)));

// Pointee type required by the async-to-LDS builtin (from hipcc diagnostic):
// 'int __attribute__((vector_size(4 * sizeof(int))))'
typedef int v4i __attribute__((vector_size(16)));
typedef __attribute__((address_space(1))) v4i gv4i;   // global
typedef __attribute__((address_space(3))) v4i lv4i;   // LDS

#if __has_builtin(__builtin_amdgcn_global_load_async_to_lds_b128)
  #define ASYNC_LDS 1
#else
  #define ASYNC_LDS 0
#endif

__global__ __launch_bounds__(BLOCK) void lac_int4_quant_rows(
    const float* __restrict__ x,
    const float* __restrict__ clip_max_p,
    const float* __restrict__ clip_min_p,
    float* __restrict__ q_out,
    float* __restrict__ scale_out)
{
    __shared__ float lds_row[ROW_D];          // 16 KiB row stage
    __shared__ float red_max[NWAVES];
    __shared__ float red_min[NWAVES];
    __shared__ float s_scale;

    const int row = blockIdx.x;
    const int t   = threadIdx.x;
    const float* __restrict__ xr = x + (size_t)row * ROW_D;

    // ---- stage the row into LDS (async DMA path on CDNA5) ----
#if ASYNC_LDS
    #pragma unroll
    for (int k = 0; k < V4PT; ++k) {
        const int i = (k * BLOCK + t) * 4;   // coalesced b128 per lane
        __builtin_amdgcn_global_load_async_to_lds_b128(
            (gv4i*)(xr + i),
            (lv4i*)(&lds_row[i]),
            /*offset=*/0, /*cpol=*/0);
    }
  #if __has_builtin(__builtin_amdgcn_s_wait_asynccnt)
    __builtin_amdgcn_s_wait_asynccnt(0);
  #else
    asm volatile("s_wait_asynccnt 0" ::: "memory");
  #endif
#else
    #pragma unroll
    for (int k = 0; k < V4PT; ++k) {
        const int i = (k * BLOCK + t) * 4;
        *(f4*)(&lds_row[i]) = *(const f4*)(xr + i);
    }
#endif
    __syncthreads();

    // ---- pull this thread's 16 values into registers; local max/min ----
    f4 vals[V4PT];
    float vmax = -FLT_MAX, vmin = FLT_MAX;
    #pragma unroll
    for (int k = 0; k < V4PT; ++k) {
        const int i = (k * BLOCK + t) * 4;
        f4 v = *(const f4*)(&lds_row[i]);
        vals[k] = v;
        vmax = fmaxf(vmax, fmaxf(fmaxf(v.x, v.y), fmaxf(v.z, v.w)));
        vmin = fminf(vmin, fminf(fminf(v.x, v.y), fminf(v.z, v.w)));
    }

    // ---- wave32 butterfly reduction, then cross-wave via LDS ----
    #pragma unroll
    for (int off = 16; off > 0; off >>= 1) {
        vmax = fmaxf(vmax, __shfl_xor(vmax, off, 32));
        vmin = fminf(vmin, __shfl_xor(vmin, off, 32));
    }
    const int wave = t >> 5;
    const int lane = t & 31;
    if (lane == 0) { red_max[wave] = vmax; red_min[wave] = vmin; }
    __syncthreads();

    if (t == 0) {
        float m = red_max[0], n = red_min[0];
        #pragma unroll
        for (int w = 1; w < NWAVES; ++w) {
            m = fmaxf(m, red_max[w]);
            n = fminf(n, red_min[w]);
        }
        // LAC: sigmoid-scaled clip, symmetric range, fp16 scale
        const float sig_max = 1.0f / (1.0f + expf(-clip_max_p[0]));
        const float sig_min = 1.0f / (1.0f + expf(-clip_min_p[0]));
        float xmax = fmaxf(m, 0.0f) * sig_max;
        float xmin = fminf(n, 0.0f) * sig_min;
        xmax = fmaxf(fabsf(xmin), xmax);
        float sc = (xmax == 0.0f) ? 1.0f : (xmax / 7.0f);   // IEEE div, like reference
        sc = __half2float(__float2half_rn(sc));             // scales.astype(float16)
        s_scale = sc;
        scale_out[row] = sc;
    }
    __syncthreads();

    // ---- quantize from registers: clamp(rint(x/scale), -8, 7) ----
    const float s = s_scale;
    float* __restrict__ qr = q_out + (size_t)row * ROW_D;
    #pragma unroll
    for (int k = 0; k < V4PT; ++k) {
        const int i = (k * BLOCK + t) * 4;
        f4 v = vals[k];
        f4 q;
        q.x = fminf(fmaxf(rintf(v.x / s), -8.0f), 7.0f);
        q.y = fminf(fmaxf(rintf(v.y / s), -8.0f), 7.0f);
        q.z = fminf(fmaxf(rintf(v.z / s), -8.0f), 7.0f);
        q.w = fminf(fmaxf(rintf(v.w / s), -8.0f), 7.0f);
        *(f4*)(qr + i) = q;
    }
}

extern "C" void kernel_launch(void* const* d_in, const int* in_sizes, int n_in,
                              void* d_out, int out_size, void* d_ws, size_t ws_size,
                              hipStream_t stream) {
    (void)n_in; (void)out_size; (void)d_ws; (void)ws_size;
    const float* x    = (const float*)d_in[0];
    const float* cmax = (const float*)d_in[1];
    const float* cmin = (const float*)d_in[2];

    const int n     = in_sizes[0];        // 2*4096*4096
    const int nrows = n / ROW_D;          // 8192

    float* q_out     = (float*)d_out;     // [n] quantized values (fp32)
    float* scale_out = q_out + n;         // [nrows] fp16-rounded scales as float

    lac_int4_quant_rows<<<nrows, BLOCK, 0, stream>>>(x, cmax, cmin, q_out, scale_out);
}